// CrystalDecoder_16260746183171
// MI455X (gfx1250) — compile-verified
//
#include <hip/hip_runtime.h>
#include <hip/hip_bf16.h>

typedef __attribute__((ext_vector_type(2))) float v2f;
typedef __attribute__((ext_vector_type(4))) float v4f;
typedef __attribute__((ext_vector_type(8))) float v8f;

// ---------------------------------------------------------------------------
// Generic fp32 WMMA GEMM:  C[M,N] = act(A[M,K] @ B[K,N] + bias[N])
// One wave32 computes a 16x16 output tile via V_WMMA_F32_16X16X4_F32.
// 4 waves per 128-thread block, striped along N.
//
// Per ISA 7.12.2 (32-bit operand layouts, wave32):
//   A (16x4):  lane L (r=L&15, c=L>>4): vgpr v holds A[r][k0 + 2*c + v]
//   B (4x16):  lane L:                  vgpr v holds B[k0 + 2*c + v][n0 + r]
//   C (16x16): vgpr i, lanes 0-15 -> row i, lanes 16-31 -> row i+8, col = r
// ---------------------------------------------------------------------------
template <bool RELU>
__global__ __launch_bounds__(128) void wmma_gemm_f32_kernel(
    const float* __restrict__ A, const float* __restrict__ B,
    const float* __restrict__ bias, float* __restrict__ C,
    int M, int N, int K)
{
  const int lane  = threadIdx.x & 31;
  const int wave  = threadIdx.x >> 5;
  const int tileN = blockIdx.x * 4 + wave;
  const int tileM = blockIdx.y;
  const int n0 = tileN * 16;
  const int m0 = tileM * 16;
  if (n0 >= N || m0 >= M) return;

  const int r = lane & 15;   // row-in-tile for A, column-in-tile for B/C
  const int c = lane >> 4;   // half-wave select (K sub-pair / M upper half)

  v8f acc = {};
  const float* __restrict__ Ap = A + (long)(m0 + r) * K + 2 * c;
  const float* __restrict__ Bp = B + (long)(2 * c) * N + n0 + r;

  for (int k0 = 0; k0 < K; k0 += 4) {
    v2f a, b;
    // contiguous pair from row-major A
    a.x = Ap[k0 + 0];
    a.y = Ap[k0 + 1];
    // two K-adjacent rows of row-major B
    const float* bp = Bp + (long)k0 * N;
    b.x = bp[0];
    b.y = bp[N];
    // 8 args: (neg_a, A, neg_b, B, c_mod, C, reuse_a, reuse_b)
    acc = __builtin_amdgcn_wmma_f32_16x16x4_f32(
        false, a, false, b, (short)0, acc, false, false);
  }

  const float bv = bias[n0 + r];
#pragma unroll
  for (int i = 0; i < 8; ++i) {
    float v = acc[i] + bv;
    if (RELU) v = fmaxf(v, 0.0f);
    C[(long)(m0 + i + 8 * c) * N + n0 + r] = v;
  }
}

// ---------------------------------------------------------------------------
// Row gather: out[node][0:128] = table[seg[node]][0:128]
// One 128-bit chunk per thread; 32 threads cover one node's 128-float row,
// so writes are fully coalesced. The 2 MB table has ~64x row reuse and must
// stay resident in L2 (192 MB), while the 134 MB output is write-once:
//   - output stores use TH=NT (non-temporal) so the write stream rinses to
//     HBM instead of evicting the hot table,
//   - segment_ids (read-once stream) uses an NT load.
// Traffic ~= the 134 MB output write -> ~6 us floor at 23.3 TB/s.
// ---------------------------------------------------------------------------
__global__ __launch_bounds__(256) void gather_rows_kernel(
    const float* __restrict__ table, const int* __restrict__ seg,
    float* __restrict__ out, int n_nodes)
{
  const long t    = (long)blockIdx.x * blockDim.x + threadIdx.x;
  const int  node = (int)(t >> 5);
  const int  q    = (int)(t & 31);
  if (node >= n_nodes) return;
  const int g = __builtin_nontemporal_load(seg + node);
  const v4f* __restrict__ src = (const v4f*)(table + (long)g * 128);
  v4f* __restrict__ dst       = (v4f*)(out + (long)node * 128);
  const v4f v = src[q];                      // RT load: keep table in L2
  __builtin_nontemporal_store(v, dst + q);   // NT store: stream out to HBM
}

// ---------------------------------------------------------------------------
// Launch: 3 graph-level GEMMs (hoisted before the segment broadcast) + gather
//
//   out = ( relu(relu(z@W_lp+b_lp) @ W1 + b1) @ W2 + b2 )[segment_ids]
//
// The linear layers are row-wise maps, so they commute with the row gather:
// compute per-graph (4096 rows) instead of per-node (262144 rows), cutting
// 155 GFLOP -> 2.4 GFLOP and leaving a pure 134 MB bandwidth problem.
// ---------------------------------------------------------------------------
extern "C" void kernel_launch(void* const* d_in, const int* in_sizes, int n_in,
                              void* d_out, int out_size, void* d_ws, size_t ws_size,
                              hipStream_t stream) {
  constexpr int LATENT = 256;
  constexpr int H      = 256;   // HIDDEN
  constexpr int TWOH   = 512;   // 2*HIDDEN
  constexpr int F      = 128;   // NODE_FEAT

  const float* z    = (const float*)d_in[0];
  const int*   seg  = (const int*)d_in[1];
  const float* W_lp = (const float*)d_in[2];
  const float* b_lp = (const float*)d_in[3];
  const float* W1   = (const float*)d_in[4];
  const float* b1   = (const float*)d_in[5];
  const float* W2   = (const float*)d_in[6];
  const float* b2   = (const float*)d_in[7];
  float* out = (float*)d_out;

  const int B       = in_sizes[0] / LATENT;  // 4096 graphs
  const int n_nodes = in_sizes[1];           // 262144 nodes

  // Workspace layout (floats): h [B,512] | hid [B,256] | small [B,128]
  float* h     = (float*)d_ws;
  float* hid   = h + (long)B * TWOH;
  float* small = hid + (long)B * H;

  dim3 blk(128, 1, 1);

  // L1: h = relu(z @ W_lp + b_lp)           [B,256] @ [256,512]
  wmma_gemm_f32_kernel<true><<<dim3(TWOH / 64, B / 16), blk, 0, stream>>>(
      z, W_lp, b_lp, h, B, TWOH, LATENT);

  // L2: hid = relu(h @ W1 + b1)             [B,512] @ [512,256]
  wmma_gemm_f32_kernel<true><<<dim3(H / 64, B / 16), blk, 0, stream>>>(
      h, W1, b1, hid, B, H, TWOH);

  // L3: small = hid @ W2 + b2               [B,256] @ [256,128]
  wmma_gemm_f32_kernel<false><<<dim3(F / 64, B / 16), blk, 0, stream>>>(
      hid, W2, b2, small, B, F, H);

  // Gather: out[node] = small[seg[node]]    (134 MB NT stream, the real cost)
  const long threads = (long)n_nodes * (F / 4);
  const int  gblocks = (int)((threads + 255) / 256);
  gather_rows_kernel<<<gblocks, 256, 0, stream>>>(small, seg, out, n_nodes);
}